// GraphConv_76141180224085
// MI455X (gfx1250) — compile-verified
//
#include <hip/hip_runtime.h>
#include <hip/hip_bf16.h>

#define F 128           // F_IN == F_OUT == 128

typedef __attribute__((ext_vector_type(2))) float v2f;
typedef __attribute__((ext_vector_type(8))) float v8f;

// ---------------------------------------------------------------------------
// Kernel 1: zero out[] (N*128) and deg[] (N)
// ---------------------------------------------------------------------------
__global__ void __launch_bounds__(256)
gcn_init(float* __restrict__ out, float* __restrict__ deg, long long total, int N) {
    long long i = (long long)blockIdx.x * 256 + threadIdx.x;
    if (i < total) out[i] = 0.0f;
    if (i < N)     deg[i] = 0.0f;
}

// ---------------------------------------------------------------------------
// Kernel 2: degree over destination rows: deg[row[e]] += 1
// ---------------------------------------------------------------------------
__global__ void __launch_bounds__(256)
gcn_degree(const int* __restrict__ ei, float* __restrict__ deg, int E) {
    int e = blockIdx.x * 256 + threadIdx.x;
    if (e < E) {
        unsafeAtomicAdd(deg + ei[e], 1.0f);   // hardware global_atomic_add_f32
    }
}

// ---------------------------------------------------------------------------
// Kernel 3: deg_inv_sqrt, including +1 for the self loop (deg >= 1 always)
// ---------------------------------------------------------------------------
__global__ void __launch_bounds__(256)
gcn_rsqrt(float* __restrict__ deg, int N) {
    int i = blockIdx.x * 256 + threadIdx.x;
    if (i < N) {
        deg[i] = rsqrtf(deg[i] + 1.0f);
    }
}

// ---------------------------------------------------------------------------
// Kernel 4: h = x @ W + b via V_WMMA_F32_16X16X4_F32 (fp32-exact matrix pipe)
// Block = 256 threads = 8 waves; each wave -> 16 rows x 128 cols of h.
// W staged in LDS K-pair interleaved so every B fragment is ONE ds_load_b64
// into an even-aligned VGPR pair (no register shuffling before the WMMA).
//   lwp[k2*128 + col] = { W[2*k2][col], W[2*k2+1][col] }
// ---------------------------------------------------------------------------
__global__ void __launch_bounds__(256)
gcn_gemm_wmma(const float* __restrict__ x, const float* __restrict__ W,
              const float* __restrict__ bias, float* __restrict__ h, int N) {
    __shared__ float2 lwp[(F / 2) * F];    // 64 KB (WGP has 320 KB LDS)

    const int t = threadIdx.x;
    // Cooperative interleaved stage: thread handles 4 consecutive cols of one
    // row-pair per step.  64 row-pairs * 32 col-quads = 2048 chunks.
    for (int i = t; i < (F / 2) * (F / 4); i += 256) {
        const int k2 = i >> 5;            // 0..63  (row pair)
        const int c4 = (i & 31) * 4;      // 0..124 (col quad)
        const float4 a = *(const float4*)(W + (size_t)(2 * k2)     * F + c4);
        const float4 b = *(const float4*)(W + (size_t)(2 * k2 + 1) * F + c4);
        float2* dst = &lwp[k2 * F + c4];  // 4 float2 = contiguous 32B in LDS
        dst[0] = make_float2(a.x, b.x);
        dst[1] = make_float2(a.y, b.y);
        dst[2] = make_float2(a.z, b.z);
        dst[3] = make_float2(a.w, b.w);
    }
    __syncthreads();

    const int wave   = t >> 5;             // 0..7
    const int lane   = t & 31;
    const int lane16 = lane & 15;
    const int hi     = lane >> 4;          // 0: lanes 0-15, 1: lanes 16-31
    const int rowbase = (blockIdx.x * 8 + wave) * 16;

    // accumulators: 8 N-tiles of 16 cols each, seeded with the bias
    v8f acc[8];
    #pragma unroll
    for (int n = 0; n < 8; ++n) {
        float bv = bias[n * 16 + lane16];
        #pragma unroll
        for (int r = 0; r < 8; ++r) acc[n][r] = bv;
    }

    // A-matrix row for this lane (clamped for the ragged tail; stores guarded)
    int arow = rowbase + lane16;
    if (arow >= N) arow = N - 1;
    const float* xr = x + (size_t)arow * F;

    const v2f* lwv = (const v2f*)lwp;

    for (int k = 0; k < F / 4; ++k) {
        // A 16x4 layout: lanes 0-15 hold K=4k,4k+1 ; lanes 16-31 hold K=4k+2,4k+3
        const int kb = 4 * k + 2 * hi;
        v2f a = *(const v2f*)(xr + kb);          // one global_load_b64
        const int kpair = 2 * k + hi;            // interleaved row-pair index
        const v2f* brow = lwv + kpair * F + lane16;
        #pragma unroll
        for (int n = 0; n < 8; ++n) {
            v2f bf = brow[n * 16];               // one ds_load_b64, aligned pair
            acc[n] = __builtin_amdgcn_wmma_f32_16x16x4_f32(
                false, a, false, bf, (short)0, acc[n], false, false);
        }
    }

    // C/D layout: VGPR r holds row M=r (lanes 0-15) and M=r+8 (lanes 16-31)
    #pragma unroll
    for (int n = 0; n < 8; ++n) {
        const int col = n * 16 + lane16;
        #pragma unroll
        for (int r = 0; r < 8; ++r) {
            const int row = rowbase + r + 8 * hi;
            if (row < N) h[(size_t)row * F + col] = acc[n][r];
        }
    }
}

// ---------------------------------------------------------------------------
// Kernel 5: per-edge gather/scale/scatter. One wave per edge; each lane owns
// 4 consecutive output features (float4 gather from L2-resident h, 4 hardware
// f32 atomics into L2-resident out).
// ---------------------------------------------------------------------------
__global__ void __launch_bounds__(256)
gcn_edges(const int* __restrict__ ei, const float* __restrict__ h,
          const float* __restrict__ dinv, float* __restrict__ out, int E) {
    const long long e = (long long)blockIdx.x * 8 + (threadIdx.x >> 5);
    if (e >= E) return;
    const int lane = threadIdx.x & 31;
    const int r = ei[e];
    const int c = ei[(long long)E + e];
    const float nrm = dinv[r] * dinv[c];

    const float4 hv = ((const float4*)(h + (size_t)c * F))[lane];
    float* o = out + (size_t)r * F + lane * 4;
    unsafeAtomicAdd(o + 0, nrm * hv.x);
    unsafeAtomicAdd(o + 1, nrm * hv.y);
    unsafeAtomicAdd(o + 2, nrm * hv.z);
    unsafeAtomicAdd(o + 3, nrm * hv.w);
}

// ---------------------------------------------------------------------------
// Kernel 6: self loops: out[i] += dinv[i]^2 * h[i].  Runs after kernel 5 on
// the same stream, so plain read-modify-write is race-free.
// ---------------------------------------------------------------------------
__global__ void __launch_bounds__(256)
gcn_selfloop(const float* __restrict__ h, const float* __restrict__ dinv,
             float* __restrict__ out, int N) {
    const int node = blockIdx.x * 8 + (threadIdx.x >> 5);
    if (node >= N) return;
    const int lane = threadIdx.x & 31;
    float s = dinv[node];
    s = s * s;
    const float4 hv = ((const float4*)(h   + (size_t)node * F))[lane];
    float4*      op = ((float4*)      (out + (size_t)node * F)) + lane;
    float4 ov = *op;
    ov.x += s * hv.x;  ov.y += s * hv.y;
    ov.z += s * hv.z;  ov.w += s * hv.w;
    *op = ov;
}

// ---------------------------------------------------------------------------
extern "C" void kernel_launch(void* const* d_in, const int* in_sizes, int n_in,
                              void* d_out, int out_size, void* d_ws, size_t ws_size,
                              hipStream_t stream) {
    const float* x    = (const float*)d_in[0];   // [N, 128]
    const int*   ei   = (const int*)  d_in[1];   // [2, E] (int32 under default JAX x64-off)
    const float* W    = (const float*)d_in[2];   // [128, 128]
    const float* bias = (const float*)d_in[3];   // [128]
    float*       out  = (float*)d_out;           // [N, 128]

    const int N = in_sizes[0] / F;
    const int E = in_sizes[1] / 2;

    float* h   = (float*)d_ws;                   // N*128 floats
    float* deg = h + (size_t)N * F;              // N floats (deg, then dinv in place)

    const long long total = (long long)N * F;

    // 1) zero out + deg
    {
        int grid = (int)((total + 255) / 256);
        gcn_init<<<grid, 256, 0, stream>>>(out, deg, total, N);
    }
    // 2) degree atomics over edges
    gcn_degree<<<(E + 255) / 256, 256, 0, stream>>>(ei, deg, E);
    // 3) dinv = rsqrt(deg + 1)
    gcn_rsqrt<<<(N + 255) / 256, 256, 0, stream>>>(deg, N);
    // 4) h = x @ W + b  (WMMA fp32, 128 rows per block)
    gcn_gemm_wmma<<<(N + 127) / 128, 256, 0, stream>>>(x, W, bias, h, N);
    // 5) edge gather/scatter (1 wave per edge, 8 edges per block)
    gcn_edges<<<(E + 7) / 8, 256, 0, stream>>>(ei, h, deg, out, E);
    // 6) self loops (1 wave per node)
    gcn_selfloop<<<(N + 7) / 8, 256, 0, stream>>>(h, deg, out, N);
}